// GAT_53790170415738
// MI455X (gfx1250) — compile-verified
//
#include <hip/hip_runtime.h>
#include <hip/hip_bf16.h>

#define NNODES 50000
#define FIN    128
#define NHEAD  4
#define CDIM   32
#define NEDGE  800000
#define ETOT   (NEDGE + NNODES)   // with self loops
#define NEG_SLOPE 0.2f

typedef __attribute__((ext_vector_type(2))) float v2f;
typedef __attribute__((ext_vector_type(8))) float v8f;

// ---------------------------------------------------------------------------
// Kernel 1: xp[n, j] = sum_k x[n,k] * W[j,k]   (fp32 WMMA 16x16x4)
// One wave computes a 16(node) x 16(feature) tile; 8 waves/block -> 128 feats.
// A (16x4): lane L, row = L&15; VGPR pair holds K=(L>=16?2:0)+{0,1} -> contiguous float2.
// B (4x16): col = L&15 (a row of W, since B = W^T); same K-pair striping.
// D (16x16): VGPR v, lane L -> row = v + 8*(L>=16), col = L&15.
// ---------------------------------------------------------------------------
__global__ __launch_bounds__(256) void gat_proj_wmma(const float* __restrict__ x,
                                                     const float* __restrict__ W,
                                                     float* __restrict__ xp) {
    const int wave = threadIdx.x >> 5;        // feature tile 0..7
    const int lane = threadIdx.x & 31;
    const int nodeBase = blockIdx.x * 16;     // 3125 blocks * 16 = 50000
    const int jBase = wave * 16;
    const int r = lane & 15;                  // A row / B column within tile
    const int kOff = (lane >> 4) * 2;         // 0 or 2

    const float* __restrict__ xrow = x + (size_t)(nodeBase + r) * FIN;
    const float* __restrict__ wrow = W + (size_t)(jBase + r) * FIN;

    v8f acc = {};
#pragma unroll
    for (int k0 = 0; k0 < FIN; k0 += 4) {
        v2f a = *(const v2f*)(xrow + k0 + kOff);
        v2f b = *(const v2f*)(wrow + k0 + kOff);
        acc = __builtin_amdgcn_wmma_f32_16x16x4_f32(
            /*neg_a=*/false, a, /*neg_b=*/false, b,
            /*c_mod=*/(short)0, acc, /*reuse_a=*/false, /*reuse_b=*/false);
    }

    float* outBase = xp + (size_t)nodeBase * FIN + jBase;
    const int hi = lane >> 4;
#pragma unroll
    for (int v = 0; v < 8; ++v) {
        const int m = v + 8 * hi;
        outBase[(size_t)m * FIN + r] = acc[v];   // 16 lanes -> 16 contiguous floats
    }
}

// ---------------------------------------------------------------------------
// Kernel 2: a_src[n,h] = sum_c xp[n,h,c]*att_src[h,c]  (and a_dst).
// One wave per node; lane c handles channel c of all 4 heads; shuffle-reduce.
// ---------------------------------------------------------------------------
__global__ __launch_bounds__(256) void gat_att_dots(const float* __restrict__ xp,
                                                    const float* __restrict__ att_src,
                                                    const float* __restrict__ att_dst,
                                                    float* __restrict__ a_src,
                                                    float* __restrict__ a_dst) {
    const int lane = threadIdx.x & 31;
    const int node = blockIdx.x * 8 + (threadIdx.x >> 5);   // 6250 blocks * 8 = 50000
    if (node >= NNODES) return;

    const float* xr = xp + (size_t)node * FIN;
    float ps[NHEAD], pd[NHEAD];
#pragma unroll
    for (int h = 0; h < NHEAD; ++h) {
        const float v = xr[h * CDIM + lane];
        ps[h] = v * att_src[h * CDIM + lane];
        pd[h] = v * att_dst[h * CDIM + lane];
    }
#pragma unroll
    for (int off = 16; off >= 1; off >>= 1) {
#pragma unroll
        for (int h = 0; h < NHEAD; ++h) {
            ps[h] += __shfl_down(ps[h], off, 32);
            pd[h] += __shfl_down(pd[h], off, 32);
        }
    }
    if (lane == 0) {
#pragma unroll
        for (int h = 0; h < NHEAD; ++h) {
            a_src[node * NHEAD + h] = ps[h];
            a_dst[node * NHEAD + h] = pd[h];
        }
    }
}

// Monotone float <-> uint mapping so atomicMax(u32) == float max.
__device__ __forceinline__ unsigned fmap(float f) {
    unsigned b = __float_as_uint(f);
    return b ^ ((b >> 31) ? 0xffffffffu : 0x80000000u);
}
__device__ __forceinline__ float funmap(unsigned u) {
    unsigned b = u ^ ((u >> 31) ? 0x80000000u : 0xffffffffu);
    return __uint_as_float(b);
}

__device__ __forceinline__ void edge_sd(int e, const int* __restrict__ ei, int& s, int& d) {
    if (e < NEDGE) { s = ei[e]; d = ei[NEDGE + e]; }
    else           { s = e - NEDGE; d = s; }       // self loop
}

// ---------------------------------------------------------------------------
// Kernel 3: per-edge leaky-relu logits + segment max via uint atomicMax.
// ---------------------------------------------------------------------------
__global__ void gat_logit_max(const int* __restrict__ ei,
                              const float* __restrict__ a_src,
                              const float* __restrict__ a_dst,
                              float* __restrict__ logits,
                              unsigned* __restrict__ segmax) {
    const int tid = blockIdx.x * blockDim.x + threadIdx.x;
    if (tid >= ETOT * NHEAD) return;
    const int e = tid >> 2, h = tid & 3;
    int s, d; edge_sd(e, ei, s, d);
    float l = a_src[s * NHEAD + h] + a_dst[d * NHEAD + h];
    l = (l > 0.0f) ? l : NEG_SLOPE * l;
    logits[tid] = l;
    atomicMax(&segmax[d * NHEAD + h], fmap(l));
}

// ---------------------------------------------------------------------------
// Kernel 4: e = exp(logit - max[dst]); denom[dst] += e  (e stored in place).
// ---------------------------------------------------------------------------
__global__ void gat_expsum(const int* __restrict__ ei,
                           const unsigned* __restrict__ segmax,
                           float* __restrict__ logits,
                           float* __restrict__ denom) {
    const int tid = blockIdx.x * blockDim.x + threadIdx.x;
    if (tid >= ETOT * NHEAD) return;
    const int e = tid >> 2, h = tid & 3;
    int s, d; edge_sd(e, ei, s, d);
    (void)s;
    const float m = funmap(segmax[d * NHEAD + h]);
    const float ex = __expf(logits[tid] - m);
    logits[tid] = ex;
    atomicAdd(&denom[d * NHEAD + h], ex);
}

// ---------------------------------------------------------------------------
// Kernel 5: out[dst, j] += (e / denom[dst]) * xp[src, j]   (128 lanes / edge)
// ---------------------------------------------------------------------------
#define EDGES_PER_BLOCK 8
__global__ __launch_bounds__(128) void gat_aggregate(const int* __restrict__ ei,
                                                     const float* __restrict__ xp,
                                                     const float* __restrict__ ew,
                                                     const float* __restrict__ denom,
                                                     float* __restrict__ out) {
    const int j = threadIdx.x;     // 0..127
    const int h = j >> 5;          // head
    const int base = blockIdx.x * EDGES_PER_BLOCK;
#pragma unroll 1
    for (int i = 0; i < EDGES_PER_BLOCK; ++i) {
        const int e = base + i;
        if (e >= ETOT) return;
        int s, d; edge_sd(e, ei, s, d);
        const float alpha = ew[e * NHEAD + h] / (denom[d * NHEAD + h] + 1e-16f);
        atomicAdd(&out[(size_t)d * FIN + j], alpha * xp[(size_t)s * FIN + j]);
    }
}

// ---------------------------------------------------------------------------
// Kernel 6: out = elu(out + bias)
// ---------------------------------------------------------------------------
__global__ void gat_bias_elu(float* __restrict__ out, const float* __restrict__ bias) {
    const int tid = blockIdx.x * blockDim.x + threadIdx.x;
    if (tid >= NNODES * FIN) return;
    const float v = out[tid] + bias[tid & (FIN - 1)];
    out[tid] = (v > 0.0f) ? v : (__expf(v) - 1.0f);
}

// ---------------------------------------------------------------------------
extern "C" void kernel_launch(void* const* d_in, const int* in_sizes, int n_in,
                              void* d_out, int out_size, void* d_ws, size_t ws_size,
                              hipStream_t stream) {
    const float* x        = (const float*)d_in[0];   // [N, 128]
    const float* W        = (const float*)d_in[1];   // [128, 128]
    const float* att_src  = (const float*)d_in[2];   // [1, 4, 32]
    const float* att_dst  = (const float*)d_in[3];   // [1, 4, 32]
    const float* bias     = (const float*)d_in[4];   // [128]
    const int*   ei       = (const int*)d_in[5];     // [2, 800000]
    float* out            = (float*)d_out;           // [N, 128]

    // Workspace layout
    char* ws = (char*)d_ws;
    float*    xp     = (float*)ws;                               ws += (size_t)NNODES * FIN * 4;      // 25.6 MB
    float*    a_src  = (float*)ws;                               ws += (size_t)NNODES * NHEAD * 4;    // 0.8 MB
    float*    a_dst  = (float*)ws;                               ws += (size_t)NNODES * NHEAD * 4;    // 0.8 MB
    float*    logits = (float*)ws;                               ws += (size_t)ETOT * NHEAD * 4;      // 13.6 MB
    float*    denom  = (float*)ws;                               ws += (size_t)NNODES * NHEAD * 4;    // 0.8 MB
    unsigned* segmax = (unsigned*)ws;                            ws += (size_t)NNODES * NHEAD * 4;    // 0.8 MB

    // Zero accumulators (graph-capture-safe async memsets)
    hipMemsetAsync(segmax, 0, (size_t)NNODES * NHEAD * 4, stream);
    hipMemsetAsync(denom,  0, (size_t)NNODES * NHEAD * 4, stream);
    hipMemsetAsync(out,    0, (size_t)NNODES * FIN * 4,   stream);

    // 1) projection GEMM via f32 WMMA
    gat_proj_wmma<<<NNODES / 16, 256, 0, stream>>>(x, W, xp);
    // 2) attention dot products
    gat_att_dots<<<NNODES / 8, 256, 0, stream>>>(xp, att_src, att_dst, a_src, a_dst);
    // 3) logits + segment max
    {
        const int total = ETOT * NHEAD;
        gat_logit_max<<<(total + 255) / 256, 256, 0, stream>>>(ei, a_src, a_dst, logits, segmax);
    }
    // 4) exp + segment sum
    {
        const int total = ETOT * NHEAD;
        gat_expsum<<<(total + 255) / 256, 256, 0, stream>>>(ei, segmax, logits, denom);
    }
    // 5) weighted scatter-add aggregation
    {
        const int blocks = (ETOT + EDGES_PER_BLOCK - 1) / EDGES_PER_BLOCK;
        gat_aggregate<<<blocks, 128, 0, stream>>>(ei, xp, logits, denom, out);
    }
    // 6) bias + ELU
    {
        const int total = NNODES * FIN;
        gat_bias_elu<<<(total + 255) / 256, 256, 0, stream>>>(out, bias);
    }
}